// GINModel_10986526343328
// MI455X (gfx1250) — compile-verified
//
#include <hip/hip_runtime.h>
#include <hip/hip_bf16.h>

#define N_NODES 100000
#define N_EDGES 1600000
#define DIM     128
#define L_LAYERS 3
#define HIDN    256
#define NLAB    6
#define BN_EPS  1e-3f
#define ROWS_PER_BLK 32   // N_NODES / 32 = 3125, exact

typedef __attribute__((ext_vector_type(2))) float v2f;
typedef __attribute__((ext_vector_type(8))) float v8f;

// D(16x16,f32) = A(16x4,f32) * B(4x16,f32) + C  -- native CDNA5 f32 WMMA
__device__ __forceinline__ v8f wmma_k4(v2f a, v2f b, v8f c) {
    return __builtin_amdgcn_wmma_f32_16x16x4_f32(
        /*neg_a=*/false, a, /*neg_b=*/false, b,
        /*c_mod=*/(short)0, c, /*reuse_a=*/false, /*reuse_b=*/false);
}

// CDNA5 async copy: global -> LDS, 16 bytes per lane, tracked by ASYNCcnt.
// vdst = LDS byte address (low 32 bits of generic shared pointer),
// vaddr = per-lane 32-bit byte offset, saddr = 64-bit SGPR base.
__device__ __forceinline__ void async_b128_to_lds(const void* gbase, unsigned goff_bytes,
                                                  void* lds_ptr) {
    unsigned ldsOff = (unsigned)(size_t)lds_ptr;
    asm volatile("global_load_async_to_lds_b128 %0, %1, %2 offset:0"
                 :: "v"(ldsOff), "v"(goff_bytes), "s"(gbase)
                 : "memory");
}
__device__ __forceinline__ void wait_async0() {
    asm volatile("s_wait_asynccnt 0" ::: "memory");
}

// ---------------------------------------------------------------------------
// agg := h   (float4 copy)
// ---------------------------------------------------------------------------
__global__ void copy4_kernel(const float4* __restrict__ in, float4* __restrict__ out, int n4) {
    int i = blockIdx.x * blockDim.x + threadIdx.x;
    if (i < n4) out[i] = in[i];
}

// out[c*R + r] = in[r*C + c]   (row-major RxC -> its transpose, row-major CxR)
__global__ void transpose_kernel(const float* __restrict__ in, float* __restrict__ out,
                                 int R, int C) {
    int idx = blockIdx.x * blockDim.x + threadIdx.x;
    if (idx < R * C) {
        int r = idx / C, c = idx - r * C;
        out[c * R + r] = in[idx];
    }
}

// ---------------------------------------------------------------------------
// Edge scatter: one wave per edge, one float4 feature chunk per lane.
// agg[dst] += h[src] * w   via native global_atomic_add_f32 (L2-resident).
// ---------------------------------------------------------------------------
__global__ void edge_scatter_kernel(const float* __restrict__ h,
                                    const long long* __restrict__ ei,
                                    const float* __restrict__ ew,
                                    float* __restrict__ agg) {
    const int wave = threadIdx.x >> 5;
    const int lane = threadIdx.x & 31;
    const int e    = blockIdx.x * 8 + wave;          // uniform per wave
    const long long s = ei[e];
    const long long d = ei[(long long)N_EDGES + e];
    const float     w = ew[e];
    const int c = lane << 2;                          // 0..124 step 4
    const float4 m = *(const float4*)(h + s * DIM + c);
    float* ap = agg + d * DIM + c;
#if defined(__HIP_DEVICE_COMPILE__)
    unsafeAtomicAdd(ap + 0, m.x * w);
    unsafeAtomicAdd(ap + 1, m.y * w);
    unsafeAtomicAdd(ap + 2, m.z * w);
    unsafeAtomicAdd(ap + 3, m.w * w);
#else
    (void)ap; (void)m;
#endif
}

// ---------------------------------------------------------------------------
// Fused GIN layer over 32 rows per block (2 row-tiles share every B fragment):
//   z1 = relu(agg @ W1 + b1);  h' = relu(BN(z1 @ W2 + b2))
// Weights pre-transposed: Wt[col][k] -> contiguous b64 B-fragment loads.
// ---------------------------------------------------------------------------
__global__ __launch_bounds__(256, 2)
void gin_layer_kernel(const float* __restrict__ Ain,
                      const float* __restrict__ Wt1, const float* __restrict__ b1,
                      const float* __restrict__ Wt2, const float* __restrict__ b2,
                      const float* __restrict__ gmm, const float* __restrict__ bta,
                      const float* __restrict__ mmean, const float* __restrict__ mvar,
                      float* __restrict__ Hout) {
    __shared__ float As[ROWS_PER_BLK][DIM + 4];
    __shared__ float Zs[ROWS_PER_BLK][DIM + 4];

    const int t    = threadIdx.x;
    const int wave = t >> 5;
    const int lane = t & 31;
    const int half = lane >> 4;    // K sub-block select (ISA f32 A/B layout)
    const int lm   = lane & 15;    // A row / B,C column within tile
    const int row0 = blockIdx.x * ROWS_PER_BLK;

    // async-stage 32x128 A tile into LDS (4 x b128 per thread)
    #pragma unroll
    for (int i = 0; i < 4; ++i) {
        const int idx = i * 256 + t;           // 0..1023 float4 slots
        const int r   = idx >> 5;              // 32 float4 per row
        const int c   = (idx & 31) << 2;
        async_b128_to_lds(Ain, (unsigned)(((size_t)(row0 + r) * DIM + c) * 4), &As[r][c]);
    }
    wait_async0();
    __syncthreads();

    const int col = wave * 16 + lm;

    // ---- GEMM1: relu(A @ W1 + b1) -> Zs ----
    v8f acc0 = {}, acc1 = {};
    #pragma unroll 4
    for (int k = 0; k < DIM; k += 4) {
        const int ka = k + half * 2;
        const v2f b  = *(const v2f*)(Wt1 + (size_t)col * DIM + ka);
        v2f a0; a0.x = As[lm][ka];      a0.y = As[lm][ka + 1];
        v2f a1; a1.x = As[16 + lm][ka]; a1.y = As[16 + lm][ka + 1];
        acc0 = wmma_k4(a0, b, acc0);
        acc1 = wmma_k4(a1, b, acc1);
    }
    const float bb1 = b1[col];
    #pragma unroll
    for (int r = 0; r < 8; ++r) {
        float v0 = acc0[r] + bb1, v1 = acc1[r] + bb1;
        Zs[r + 8 * half][col]      = v0 > 0.f ? v0 : 0.f;
        Zs[16 + r + 8 * half][col] = v1 > 0.f ? v1 : 0.f;
    }
    __syncthreads();

    // ---- GEMM2 + BN + relu -> global ----
    v8f acc2 = {}, acc3 = {};
    #pragma unroll 4
    for (int k = 0; k < DIM; k += 4) {
        const int ka = k + half * 2;
        const v2f b  = *(const v2f*)(Wt2 + (size_t)col * DIM + ka);
        v2f a0; a0.x = Zs[lm][ka];      a0.y = Zs[lm][ka + 1];
        v2f a1; a1.x = Zs[16 + lm][ka]; a1.y = Zs[16 + lm][ka + 1];
        acc2 = wmma_k4(a0, b, acc2);
        acc3 = wmma_k4(a1, b, acc3);
    }
    const float sc  = gmm[col] * rsqrtf(mvar[col] + BN_EPS);
    const float sh  = bta[col] - mmean[col] * sc;
    const float bb2 = b2[col];
    #pragma unroll
    for (int r = 0; r < 8; ++r) {
        float v0 = (acc2[r] + bb2) * sc + sh;
        float v1 = (acc3[r] + bb2) * sc + sh;
        v0 = v0 > 0.f ? v0 : 0.f;
        v1 = v1 > 0.f ? v1 : 0.f;
        Hout[(size_t)(row0 + r + 8 * half) * DIM + col]      = v0;
        Hout[(size_t)(row0 + 16 + r + 8 * half) * DIM + col] = v1;
    }
}

// ---------------------------------------------------------------------------
// Head: z = relu(h@W3+b3) [32x256 in LDS], out = z@W4+b4 [32x6]
// ---------------------------------------------------------------------------
__global__ __launch_bounds__(256, 2)
void head_kernel(const float* __restrict__ Hin,
                 const float* __restrict__ Wt3, const float* __restrict__ b3,
                 const float* __restrict__ Wt4, const float* __restrict__ b4,
                 float* __restrict__ Out) {
    __shared__ float As[ROWS_PER_BLK][DIM + 4];
    __shared__ float Zs[ROWS_PER_BLK][HIDN + 4];

    const int t    = threadIdx.x;
    const int wave = t >> 5;
    const int lane = t & 31;
    const int half = lane >> 4;
    const int lm   = lane & 15;
    const int row0 = blockIdx.x * ROWS_PER_BLK;

    #pragma unroll
    for (int i = 0; i < 4; ++i) {
        const int idx = i * 256 + t;
        const int r   = idx >> 5;
        const int c   = (idx & 31) << 2;
        async_b128_to_lds(Hin, (unsigned)(((size_t)(row0 + r) * DIM + c) * 4), &As[r][c]);
    }
    wait_async0();
    __syncthreads();

    // each wave computes two 16-col tiles x two 16-row tiles of the hidden layer
    #pragma unroll
    for (int tile = 0; tile < 2; ++tile) {
        const int col = (wave * 2 + tile) * 16 + lm;
        v8f acc0 = {}, acc1 = {};
        #pragma unroll 4
        for (int k = 0; k < DIM; k += 4) {
            const int ka = k + half * 2;
            const v2f b  = *(const v2f*)(Wt3 + (size_t)col * DIM + ka);
            v2f a0; a0.x = As[lm][ka];      a0.y = As[lm][ka + 1];
            v2f a1; a1.x = As[16 + lm][ka]; a1.y = As[16 + lm][ka + 1];
            acc0 = wmma_k4(a0, b, acc0);
            acc1 = wmma_k4(a1, b, acc1);
        }
        const float bb = b3[col];
        #pragma unroll
        for (int r = 0; r < 8; ++r) {
            float v0 = acc0[r] + bb, v1 = acc1[r] + bb;
            Zs[r + 8 * half][col]      = v0 > 0.f ? v0 : 0.f;
            Zs[16 + r + 8 * half][col] = v1 > 0.f ? v1 : 0.f;
        }
    }
    __syncthreads();

    // final 256 -> 6 projection: wave 0 -> row-tile 0, wave 1 -> row-tile 1
    if (wave < 2) {
        const int rbase = wave * 16;
        v8f acc = {};
        for (int k = 0; k < HIDN; k += 4) {
            const int ka = k + half * 2;
            v2f a; a.x = Zs[rbase + lm][ka]; a.y = Zs[rbase + lm][ka + 1];
            v2f b;
            if (lm < NLAB) b = *(const v2f*)(Wt4 + (size_t)lm * HIDN + ka);
            else           { b.x = 0.f; b.y = 0.f; }
            acc = wmma_k4(a, b, acc);
        }
        if (lm < NLAB) {
            const float bb = b4[lm];
            #pragma unroll
            for (int r = 0; r < 8; ++r)
                Out[(size_t)(row0 + rbase + r + 8 * half) * NLAB + lm] = acc[r] + bb;
        }
    }
}

// ---------------------------------------------------------------------------
extern "C" void kernel_launch(void* const* d_in, const int* in_sizes, int n_in,
                              void* d_out, int out_size, void* d_ws, size_t ws_size,
                              hipStream_t stream) {
    const float*      x     = (const float*)d_in[0];
    const long long*  ei    = (const long long*)d_in[1];
    const float*      ew    = (const float*)d_in[2];
    const float*      W1    = (const float*)d_in[3];
    const float*      b1    = (const float*)d_in[4];
    const float*      W2    = (const float*)d_in[5];
    const float*      b2    = (const float*)d_in[6];
    const float*      gmm   = (const float*)d_in[7];
    const float*      bta   = (const float*)d_in[8];
    const float*      mmean = (const float*)d_in[9];
    const float*      mvar  = (const float*)d_in[10];
    const float*      W3    = (const float*)d_in[11];
    const float*      b3    = (const float*)d_in[12];
    const float*      W4    = (const float*)d_in[13];
    const float*      b4    = (const float*)d_in[14];
    float*            out   = (float*)d_out;

    const size_t nd = (size_t)N_NODES * DIM;
    float* hA  = (float*)d_ws;
    float* hB  = hA + nd;
    float* agg = hB + nd;
    float* wt1 = agg + nd;                       // 3 * D * D
    float* wt2 = wt1 + (size_t)L_LAYERS * DIM * DIM;
    float* wt3 = wt2 + (size_t)L_LAYERS * DIM * DIM;   // D * HIDN
    float* wt4 = wt3 + (size_t)DIM * HIDN;             // HIDN * NLAB
    float* bufs[2] = { hA, hB };

    // one-time (per launch) weight transposes: B fragments become contiguous b64
    const int tpb = 256;
    for (int l = 0; l < L_LAYERS; ++l) {
        transpose_kernel<<<(DIM * DIM + tpb - 1) / tpb, tpb, 0, stream>>>(
            W1 + (size_t)l * DIM * DIM, wt1 + (size_t)l * DIM * DIM, DIM, DIM);
        transpose_kernel<<<(DIM * DIM + tpb - 1) / tpb, tpb, 0, stream>>>(
            W2 + (size_t)l * DIM * DIM, wt2 + (size_t)l * DIM * DIM, DIM, DIM);
    }
    transpose_kernel<<<(DIM * HIDN + tpb - 1) / tpb, tpb, 0, stream>>>(W3, wt3, DIM, HIDN);
    transpose_kernel<<<(HIDN * NLAB + tpb - 1) / tpb, tpb, 0, stream>>>(W4, wt4, HIDN, NLAB);

    const int n4          = (int)(nd / 4);
    const int copy_blocks = (n4 + tpb - 1) / tpb;
    const int edge_blocks = N_EDGES / 8;                 // one wave per edge
    const int tile_blocks = N_NODES / ROWS_PER_BLK;      // 3125, exact

    const float* hin = x;
    for (int l = 0; l < L_LAYERS; ++l) {
        copy4_kernel<<<copy_blocks, tpb, 0, stream>>>((const float4*)hin, (float4*)agg, n4);
        edge_scatter_kernel<<<edge_blocks, tpb, 0, stream>>>(hin, ei, ew, agg);
        float* hout = bufs[l & 1];
        gin_layer_kernel<<<tile_blocks, tpb, 0, stream>>>(
            agg,
            wt1 + (size_t)l * DIM * DIM, b1 + l * DIM,
            wt2 + (size_t)l * DIM * DIM, b2 + l * DIM,
            gmm + l * DIM, bta + l * DIM, mmean + l * DIM, mvar + l * DIM,
            hout);
        hin = hout;
    }
    head_kernel<<<tile_blocks, tpb, 0, stream>>>(hin, wt3, b3, wt4, b4, out);
}